// StagVI_NodeClassification_RC_65000035058538
// MI455X (gfx1250) — compile-verified
//
#include <hip/hip_runtime.h>
#include <math.h>

typedef __attribute__((ext_vector_type(2))) float v2f;
typedef __attribute__((ext_vector_type(8))) float v8f;

#define F 64
#define WSTRIDE 68   // padded LDS stride for transposed W (conflict-free b64 reads)

__device__ __forceinline__ void atomAddF32(float* p, float v) {
    // Guaranteed hardware global_atomic_add_f32 (no CAS expansion).
    unsafeAtomicAdd(p, v);
}

// sigma = exp(log_sigma) for both layers (128 values total)
__global__ void k_sigma(const float* __restrict__ ls0, const float* __restrict__ ls1,
                        float* __restrict__ sig) {
    int i = threadIdx.x;
    if (i < 64)        sig[i] = __expf(ls0[i]);
    else if (i < 128)  sig[i] = __expf(ls1[i - 64]);
}

// degree accumulation via fp32 atomics (deg buffers pre-zeroed)
__global__ void k_deg(const int* __restrict__ src, const int* __restrict__ dst,
                      float* __restrict__ deg_out, float* __restrict__ deg_in, int E) {
    int e = blockIdx.x * blockDim.x + threadIdx.x;
    if (e < E) {
        atomAddF32(&deg_out[src[e]], 1.0f);
        atomAddF32(&deg_in[dst[e]], 1.0f);
    }
}

// in-place deg -> norm : deg>0 ? rsqrt(deg) : 0   (deg is integral, so max(deg,1)==deg)
__global__ void k_norm(float* __restrict__ deg, int n2) {
    int i = blockIdx.x * blockDim.x + threadIdx.x;
    if (i < n2) {
        float d = deg[i];
        deg[i] = (d > 0.0f) ? rsqrtf(d) : 0.0f;
    }
}

// h[n][f] = x[n][f] * norm_out[n], float4 per thread (16 threads/node)
__global__ void k_scale_x(const float* __restrict__ x, const float* __restrict__ norm_out,
                          float* __restrict__ h, int nThreads) {
    int t = blockIdx.x * blockDim.x + threadIdx.x;
    if (t >= nThreads) return;
    int node = t >> 4;
    int q = (t & 15) * 4;
    float s = norm_out[node];
    float4 v = *reinterpret_cast<const float4*>(x + (size_t)node * F + q);
    float4 o;
    o.x = v.x * s; o.y = v.y * s; o.z = v.z * s; o.w = v.w * s;
    *reinterpret_cast<float4*>(h + (size_t)node * F + q) = o;
}

// edge message + scatter: 16 threads per edge, 4 features each (float4 path)
__global__ void k_edge(const float* __restrict__ h, const float* __restrict__ eps,
                       const float* __restrict__ mu, const float* __restrict__ sig,
                       const int* __restrict__ src, const int* __restrict__ dst,
                       float* __restrict__ agg, int E) {
    long long t = (long long)blockIdx.x * blockDim.x + threadIdx.x;
    int e = (int)(t >> 4);
    if (e >= E) return;
    int q = (int)(t & 15) * 4;
    int s = src[e];
    int d = dst[e];
    float4 ev = *reinterpret_cast<const float4*>(eps + (size_t)e * F + q);
    float4 hv = *reinterpret_cast<const float4*>(h + (size_t)s * F + q);
    float4 mv = *reinterpret_cast<const float4*>(mu + q);
    float4 gv = *reinterpret_cast<const float4*>(sig + q);
    float m0 = hv.x * fmaf(gv.x, ev.x, mv.x);
    float m1 = hv.y * fmaf(gv.y, ev.y, mv.y);
    float m2 = hv.z * fmaf(gv.z, ev.z, mv.z);
    float m3 = hv.w * fmaf(gv.w, ev.w, mv.w);
    float* ap = agg + (size_t)d * F + q;
    atomAddF32(ap + 0, m0);
    atomAddF32(ap + 1, m1);
    atomAddF32(ap + 2, m2);
    atomAddF32(ap + 3, m3);
}

// WMMA GEMM: out_tile[16 x 64] = (agg*norm_in)[16 x 64] @ W[64 x 64] (+bias, epilogue)
// One wave per 16-node tile; 64 x v_wmma_f32_16x16x4_f32 per tile.
// A layout (32-bit 16x4): lanes 0-15 hold K={0,1}, lanes 16-31 hold K={2,3}, M=lane&15.
// B layout (32-bit 4x16): lanes 0-15 hold rows K={0,1}, lanes 16-31 rows K={2,3}, N=lane&15.
// C/D layout: VGPR i -> row M=i (lanes 0-15) / M=8+i (lanes 16-31), N=lane&15 per col-tile.
// W is staged transposed in LDS (Wl[n*WSTRIDE + k]) so each B pair is one ds_load_b64,
// bank-conflict-free across the wave.
template <int FINAL>
__global__ void k_gemm(const float* __restrict__ agg, const float* __restrict__ norm_in,
                       const float* __restrict__ norm_out, const float* __restrict__ W,
                       const float* __restrict__ bias, float* __restrict__ out,
                       int nTiles, int N) {
    __shared__ float Wl[64 * WSTRIDE];
    __shared__ float bl[64];
    for (int i = threadIdx.x; i < 64 * 64; i += blockDim.x) {
        int k = i >> 6;
        int n = i & 63;
        Wl[n * WSTRIDE + k] = W[i];     // transpose: Wl[n][k] = W[k][n]
    }
    if (threadIdx.x < 64) bl[threadIdx.x] = bias[threadIdx.x];
    __syncthreads();

    int wave = threadIdx.x >> 5;
    int lane = threadIdx.x & 31;
    int tile = blockIdx.x * (blockDim.x >> 5) + wave;
    if (tile >= nTiles) return;                 // uniform per-wave exit; EXEC stays full

    int half = lane >> 4;                       // 0: K {0,1}, 1: K {2,3}
    int ln   = lane & 15;
    int rowNode = tile * 16 + ln;
    if (rowNode > N - 1) rowNode = N - 1;       // clamp (stores are guarded)
    float ni = norm_in[rowNode];
    const float2* ap = reinterpret_cast<const float2*>(agg + (size_t)rowNode * F);

    v8f acc[4] = {};
#pragma unroll
    for (int kb = 0; kb < 16; ++kb) {
        float2 av = ap[2 * kb + half];          // K = 4*kb + 2*half .. +1
        v2f a;
        a.x = av.x * ni;
        a.y = av.y * ni;
        int krow = 4 * kb + 2 * half;           // even -> 8B-aligned LDS b64 load
#pragma unroll
        for (int ct = 0; ct < 4; ++ct) {
            v2f b = *reinterpret_cast<const v2f*>(&Wl[(ct * 16 + ln) * WSTRIDE + krow]);
            acc[ct] = __builtin_amdgcn_wmma_f32_16x16x4_f32(
                false, a, false, b, (short)0, acc[ct], false, false);
        }
    }

    if (FINAL == 0) {
        // + bias, ReLU, pre-scale by next layer's norm_out
#pragma unroll
        for (int i = 0; i < 8; ++i) {
            int m = half * 8 + i;
            int node = tile * 16 + m;
            if (node < N) {
                float no = norm_out[node];
#pragma unroll
                for (int ct = 0; ct < 4; ++ct) {
                    float v = acc[ct][i] + bl[ct * 16 + ln];
                    v = v > 0.0f ? v : 0.0f;
                    out[(size_t)node * F + ct * 16 + ln] = v * no;
                }
            }
        }
    } else {
        // + bias, then row-wise softmax over 64 features, fully in-wave.
        // Row M's 64 values live in {acc[ct][i] : ct 0..3} across the 16-lane group.
#pragma unroll
        for (int i = 0; i < 8; ++i) {
            float r0 = acc[0][i] + bl[ln];
            float r1 = acc[1][i] + bl[16 + ln];
            float r2 = acc[2][i] + bl[32 + ln];
            float r3 = acc[3][i] + bl[48 + ln];
            float mx = fmaxf(fmaxf(r0, r1), fmaxf(r2, r3));
            mx = fmaxf(mx, __shfl_xor(mx, 1, 32));
            mx = fmaxf(mx, __shfl_xor(mx, 2, 32));
            mx = fmaxf(mx, __shfl_xor(mx, 4, 32));
            mx = fmaxf(mx, __shfl_xor(mx, 8, 32));  // xor 16 skipped: halves are distinct rows
            float e0 = __expf(r0 - mx);
            float e1 = __expf(r1 - mx);
            float e2 = __expf(r2 - mx);
            float e3 = __expf(r3 - mx);
            float s = e0 + e1 + e2 + e3;
            s += __shfl_xor(s, 1, 32);
            s += __shfl_xor(s, 2, 32);
            s += __shfl_xor(s, 4, 32);
            s += __shfl_xor(s, 8, 32);
            float inv = 1.0f / s;
            int node = tile * 16 + half * 8 + i;
            if (node < N) {
                float* op = out + (size_t)node * F;
                op[ln]      = e0 * inv;
                op[16 + ln] = e1 * inv;
                op[32 + ln] = e2 * inv;
                op[48 + ln] = e3 * inv;
            }
        }
    }
}

extern "C" void kernel_launch(void* const* d_in, const int* in_sizes, int n_in,
                              void* d_out, int out_size, void* d_ws, size_t ws_size,
                              hipStream_t stream) {
    const float* x     = (const float*)d_in[0];
    const int*   src   = (const int*)d_in[1];
    const int*   dst   = (const int*)d_in[2];
    const float* a_mu0 = (const float*)d_in[3];
    const float* a_ls0 = (const float*)d_in[4];
    const float* a_mu1 = (const float*)d_in[5];
    const float* a_ls1 = (const float*)d_in[6];
    const float* W0    = (const float*)d_in[7];
    const float* b0    = (const float*)d_in[8];
    const float* W1    = (const float*)d_in[9];
    const float* b1    = (const float*)d_in[10];
    const float* eps0  = (const float*)d_in[11];
    const float* eps1  = (const float*)d_in[12];
    float* out = (float*)d_out;

    const int N = in_sizes[0] / F;   // 100000
    const int E = in_sizes[1];       // 1600000

    // workspace layout (floats): [deg_out|deg_in : 2N][sig : 128][hbuf : N*F][agg : N*F]
    float* ws   = (float*)d_ws;
    float* deg  = ws;
    float* sig  = ws + 2 * (size_t)N;
    float* hbuf = sig + 128;
    float* agg  = hbuf + (size_t)N * F;
    float* norm_out_arr = deg;
    float* norm_in_arr  = deg + N;

    hipMemsetAsync(deg, 0, 2 * (size_t)N * sizeof(float), stream);
    hipMemsetAsync(agg, 0, (size_t)N * F * sizeof(float), stream);

    k_sigma<<<1, 128, 0, stream>>>(a_ls0, a_ls1, sig);
    k_deg<<<(E + 255) / 256, 256, 0, stream>>>(src, dst, norm_out_arr, norm_in_arr, E);
    k_norm<<<(2 * N + 255) / 256, 256, 0, stream>>>(deg, 2 * N);
    k_scale_x<<<(N * 16 + 255) / 256, 256, 0, stream>>>(x, norm_out_arr, hbuf, N * 16);

    long long tE = (long long)E * 16;
    int edgeBlocks = (int)((tE + 255) / 256);
    int nTiles = (N + 15) / 16;
    int gemmBlocks = (nTiles + 7) / 8;   // 8 waves (tiles) per 256-thread block

    // Layer 0: edges -> agg -> GEMM(+b0, ReLU, *norm_out) -> hbuf
    k_edge<<<edgeBlocks, 256, 0, stream>>>(hbuf, eps0, a_mu0, sig, src, dst, agg, E);
    k_gemm<0><<<gemmBlocks, 256, 0, stream>>>(agg, norm_in_arr, norm_out_arr, W0, b0,
                                              hbuf, nTiles, N);

    // Layer 1: re-zero agg, edges -> agg -> GEMM(+b1, softmax) -> out
    hipMemsetAsync(agg, 0, (size_t)N * F * sizeof(float), stream);
    k_edge<<<edgeBlocks, 256, 0, stream>>>(hbuf, eps1, a_mu1, sig + 64, src, dst, agg, E);
    k_gemm<1><<<gemmBlocks, 256, 0, stream>>>(agg, norm_in_arr, norm_out_arr, W1, b1,
                                              out, nTiles, N);
}